// TopicAttention_16681652978157
// MI455X (gfx1250) — compile-verified
//
#include <hip/hip_runtime.h>
#include <math.h>

// Problem constants (match reference)
#define BB 128     // batch
#define TT 512     // num_topics
#define VT 2048    // topic_vocab
#define EE 1024    // enc_hid
#define DD 1024    // dec_hid

// Score-kernel tiling
#define SBW 8      // batches per block
#define STW 32     // topics per block

typedef __attribute__((ext_vector_type(2))) float v2f;
typedef __attribute__((ext_vector_type(8))) float v8f;

__device__ __forceinline__ float fast_tanh(float x) {
#if __has_builtin(__builtin_amdgcn_tanhf)
    return __builtin_amdgcn_tanhf(x);   // v_tanh_f32 (gfx1250 TRANS op)
#else
    return tanhf(x);
#endif
}

// ---------------------------------------------------------------------------
// Register-blocked f32 WMMA GEMM: each wave owns an (MR*16) x (NR*16) tile.
// Fragment layout (wave32, V_WMMA_F32_16X16X4_F32):
//   A 16x4: lane<16 -> {A[row=lane%16, k], [.., k+1]}; lane>=16 -> {k+2, k+3}
//   B 4x16: same K striping per lane-half, col = lane%16
// ---------------------------------------------------------------------------
template <int MR, int NR>
__device__ __forceinline__ void gemm_accum(const float* __restrict__ A, int lda, int row0,
                                           const float* __restrict__ Bm, int ldb, int col0,
                                           int K, v8f (&acc)[MR][NR]) {
    const int lane = threadIdx.x & 31;
    const int half = lane >> 4;
    const int l16  = lane & 15;

#pragma unroll 2
    for (int k = 0; k < K; k += 4) {
        const int kk = k + 2 * half;
        v2f a[MR];
        v2f b[NR];
#pragma unroll
        for (int r = 0; r < MR; ++r) {
            const float* ap = A + (size_t)(row0 + 16 * r + l16) * lda + kk;
            a[r].x = ap[0];
            a[r].y = ap[1];
        }
#pragma unroll
        for (int c = 0; c < NR; ++c) {
            const float* bp = Bm + (size_t)kk * ldb + col0 + 16 * c + l16;
            b[c].x = bp[0];
            b[c].y = bp[ldb];
        }
#pragma unroll
        for (int r = 0; r < MR; ++r)
#pragma unroll
            for (int c = 0; c < NR; ++c)
                acc[r][c] = __builtin_amdgcn_wmma_f32_16x16x4_f32(
                    false, a[r], false, b[c], (short)0, acc[r][c], false, false);
    }
}

template <int MR, int NR>
__device__ __forceinline__ void store_tiles(float* __restrict__ C, int ldc, int row0, int col0,
                                            v8f (&acc)[MR][NR]) {
    const int lane = threadIdx.x & 31;
    const int half = lane >> 4;
    const int l16  = lane & 15;
#pragma unroll
    for (int r = 0; r < MR; ++r)
#pragma unroll
        for (int c = 0; c < NR; ++c)
#pragma unroll
            for (int i = 0; i < 8; ++i)
                C[(size_t)(row0 + 16 * r + i + 8 * half) * ldc + col0 + 16 * c + l16] = acc[r][c][i];
}

// he[b,d] = (hidden @ W_h)[b,d] + (enc @ W_e)[b,d] + bias[d]
// 1x2 tile blocking (16x32 per wave); grid (B/16, D/32), block 32.
__global__ void __launch_bounds__(32) ta_gemm_he(const float* __restrict__ hidden,
                                                 const float* __restrict__ enc,
                                                 const float* __restrict__ W_h,
                                                 const float* __restrict__ W_e,
                                                 const float* __restrict__ bias,
                                                 float* __restrict__ he) {
    const int row0 = blockIdx.x * 16;
    const int col0 = blockIdx.y * 32;
    const int l16  = threadIdx.x & 15;

    v8f acc[1][2];
#pragma unroll
    for (int c = 0; c < 2; ++c) {
        const float bv = bias[col0 + 16 * c + l16];   // per-column bias broadcast
#pragma unroll
        for (int i = 0; i < 8; ++i) acc[0][c][i] = bv;
    }

    gemm_accum<1, 2>(hidden, DD, row0, W_h, DD, col0, DD, acc);
    gemm_accum<1, 2>(enc,    EE, row0, W_e, DD, col0, EE, acc);

    store_tiles<1, 2>(he, DD, row0, col0, acc);
}

// tp = topic_dict @ W_t   [T, D], K = VT
// 2x2 tile blocking (32x32 per wave); grid (T/32, D/32), block 32.
__global__ void __launch_bounds__(32) ta_gemm_t(const float* __restrict__ topic,
                                                const float* __restrict__ W_t,
                                                float* __restrict__ tp) {
    const int row0 = blockIdx.x * 32;
    const int col0 = blockIdx.y * 32;

    v8f acc[2][2];
#pragma unroll
    for (int r = 0; r < 2; ++r)
#pragma unroll
        for (int c = 0; c < 2; ++c)
#pragma unroll
            for (int i = 0; i < 8; ++i) acc[r][c][i] = 0.0f;

    gemm_accum<2, 2>(topic, VT, row0, W_t, DD, col0, VT, acc);

    store_tiles<2, 2>(tp, DD, row0, col0, acc);
}

// ---------------------------------------------------------------------------
// scores[b,t] = sum_d tanh(he[b,d] + tp[t,d]) * v[d]
// Tile: SBW=8 batches x STW=32 topics per block; grid (T/32, B/8) = 16x16.
// tp row held in registers (8x float4) and reused across 8 batches (he in LDS);
// v hoisted to registers. Cuts L2 tp traffic 8x vs per-batch blocking.
// ---------------------------------------------------------------------------
__global__ void __launch_bounds__(256) ta_scores(const float* __restrict__ he,
                                                 const float* __restrict__ tp,
                                                 const float* __restrict__ v,
                                                 float* __restrict__ scores) {
    __shared__ float s_he[SBW][DD];   // 32 KB
    __shared__ float s_v[DD];         // 4 KB

    const int t0   = blockIdx.x * STW;
    const int b0   = blockIdx.y * SBW;
    const int tid  = threadIdx.x;
    const int wave = tid >> 5;
    const int lane = tid & 31;

    for (int i = tid; i < SBW * DD; i += 256) {
        const int bb = i >> 10;          // / DD
        const int d  = i & (DD - 1);     // % DD
        s_he[bb][d] = he[(size_t)(b0 + bb) * DD + d];
    }
    for (int d = tid; d < DD; d += 256) s_v[d] = v[d];
    __syncthreads();

    // hoist v into registers: lane owns d = 4*(lane + 32*i) .. +3
    float4 vv[8];
    const float4* s_v4 = (const float4*)s_v;
#pragma unroll
    for (int i = 0; i < 8; ++i) vv[i] = s_v4[lane + 32 * i];

#pragma unroll
    for (int ti = 0; ti < STW / 8; ++ti) {   // 4 topic rows per wave
        const int t = t0 + 8 * ti + wave;
        const float4* tpr = (const float4*)(tp + (size_t)t * DD);

        // prefetch next topic row for this wave while we crunch this one
        if (ti + 1 < STW / 8)
            __builtin_prefetch(tp + (size_t)(t + 8) * DD + 4 * lane, 0, 0);

        float4 tv[8];
#pragma unroll
        for (int i = 0; i < 8; ++i) tv[i] = tpr[lane + 32 * i];

        float acc[SBW];
#pragma unroll
        for (int bb = 0; bb < SBW; ++bb) {
            const float4* hv4 = (const float4*)s_he[bb];
            float a = 0.0f;
#pragma unroll
            for (int i = 0; i < 8; ++i) {
                const float4 hv = hv4[lane + 32 * i];
                a += fast_tanh(tv[i].x + hv.x) * vv[i].x;
                a += fast_tanh(tv[i].y + hv.y) * vv[i].y;
                a += fast_tanh(tv[i].z + hv.z) * vv[i].z;
                a += fast_tanh(tv[i].w + hv.w) * vv[i].w;
            }
            acc[bb] = a;
        }
#pragma unroll
        for (int bb = 0; bb < SBW; ++bb) {
            float a = acc[bb];
#pragma unroll
            for (int off = 16; off > 0; off >>= 1) a += __shfl_down(a, off, 32);
            if (lane == 0) scores[(size_t)(b0 + bb) * TT + t] = a;
        }
    }
}

// Row softmax over T=512; one block per batch row. out shape [B, 1, T].
__global__ void __launch_bounds__(256) ta_softmax(const float* __restrict__ scores,
                                                  float* __restrict__ out) {
    __shared__ float s[TT];
    __shared__ float red[8];

    const int b    = blockIdx.x;
    const int tid  = threadIdx.x;
    const int wave = tid >> 5;
    const int lane = tid & 31;

    for (int t = tid; t < TT; t += 256) s[t] = scores[(size_t)b * TT + t];
    __syncthreads();

    float m = -INFINITY;
    for (int t = tid; t < TT; t += 256) m = fmaxf(m, s[t]);
#pragma unroll
    for (int off = 16; off > 0; off >>= 1) m = fmaxf(m, __shfl_down(m, off, 32));
    if (lane == 0) red[wave] = m;
    __syncthreads();
    if (tid == 0) {
        float mm = red[0];
#pragma unroll
        for (int w = 1; w < 8; ++w) mm = fmaxf(mm, red[w]);
        red[0] = mm;
    }
    __syncthreads();
    m = red[0];
    __syncthreads();

    float sum = 0.0f;
    for (int t = tid; t < TT; t += 256) {
        const float e = __expf(s[t] - m);
        s[t] = e;
        sum += e;
    }
#pragma unroll
    for (int off = 16; off > 0; off >>= 1) sum += __shfl_down(sum, off, 32);
    __syncthreads();
    if (lane == 0) red[wave] = sum;
    __syncthreads();
    if (tid == 0) {
        float ss = 0.0f;
#pragma unroll
        for (int w = 0; w < 8; ++w) ss += red[w];
        red[0] = ss;
    }
    __syncthreads();
    const float inv = 1.0f / red[0];

    for (int t = tid; t < TT; t += 256)
        out[(size_t)b * TT + t] = s[t] * inv;
}

extern "C" void kernel_launch(void* const* d_in, const int* in_sizes, int n_in,
                              void* d_out, int out_size, void* d_ws, size_t ws_size,
                              hipStream_t stream) {
    (void)in_sizes; (void)n_in; (void)out_size; (void)ws_size;

    const float* hidden = (const float*)d_in[0];  // [B, D]
    const float* topic  = (const float*)d_in[1];  // [T, VT]
    const float* enc    = (const float*)d_in[2];  // [B, E]
    const float* W_h    = (const float*)d_in[3];  // [D, D]
    const float* W_t    = (const float*)d_in[4];  // [VT, D]
    const float* W_e    = (const float*)d_in[5];  // [E, D]
    const float* bias   = (const float*)d_in[6];  // [D]
    const float* v      = (const float*)d_in[7];  // [D]
    float* out = (float*)d_out;                   // [B, 1, T]

    float* he     = (float*)d_ws;                    // [B, D]
    float* tp     = he + (size_t)BB * DD;            // [T, D]
    float* scores = tp + (size_t)TT * DD;            // [B, T]

    dim3 g_he(BB / 16, DD / 32);          // 8 x 32 waves, 16x32 tiles
    ta_gemm_he<<<g_he, 32, 0, stream>>>(hidden, enc, W_h, W_e, bias, he);

    dim3 g_t(TT / 32, DD / 32);           // 16 x 32 waves, 32x32 tiles
    ta_gemm_t<<<g_t, 32, 0, stream>>>(topic, W_t, tp);

    dim3 g_s(TT / STW, BB / SBW);         // 16 x 16 blocks
    ta_scores<<<g_s, 256, 0, stream>>>(he, tp, v, scores);

    ta_softmax<<<BB, 256, 0, stream>>>(scores, out);
}